// AttentionModel_66649302499557
// MI455X (gfx1250) — compile-verified
//
#include <hip/hip_runtime.h>
#include <math.h>

typedef __attribute__((ext_vector_type(2))) float v2f;
typedef __attribute__((ext_vector_type(4))) float v4f;
typedef __attribute__((ext_vector_type(8))) float v8f;

#define H 1024
#define B 64
#define S 2048

// ---------------------------------------------------------------------------
// Kernel 1: v = hidden @ W   (64x1024 = 64x1024 * 1024x1024, contraction over o)
//   v[b,h] = sum_o hidden[b,o] * W[o,h]
// A = hidden (M=64, K=1024, row-major), Bmat = W (K=1024, N=1024, row-major).
// One wave per 16x16 tile of v, using V_WMMA_F32_16X16X4_F32 (f32-exact path).
// 4 M-tiles x 64 N-tiles = 256 waves; each wave runs K=1024 in steps of 4.
// ---------------------------------------------------------------------------
__global__ void __launch_bounds__(128)
gemm_vW_wmma(const float* __restrict__ hidden,
             const float* __restrict__ W,
             float* __restrict__ v) {
  const int lane = threadIdx.x & 31;
  const int wave = threadIdx.x >> 5;
  const int tile = blockIdx.x * 4 + wave;   // 0..255
  const int tm   = tile >> 6;               // 0..3   (b tile)
  const int tn   = tile & 63;               // 0..63  (h tile)
  const int lo16 = lane & 15;
  const int half = lane >> 4;               // 0: K=0,1  1: K=2,3 (per ISA 16x4 layout)
  const int koff = half << 1;

  const float* Arow = hidden + (size_t)(tm * 16 + lo16) * H;   // A[M=lo16, k]
  const float* Bcol = W + (tn * 16 + lo16);                    // B[k, N=lo16]

  v8f c = {};
  for (int kk = 0; kk < H; kk += 4) {
    // A 16x4: lane<16 -> (K=kk+0, kk+1), lane>=16 -> (K=kk+2, kk+3), contiguous pair
    v2f a = *(const v2f*)(Arow + kk + koff);
    // B 4x16: mirrored striping, rows kk+koff and kk+koff+1 of W
    v2f bm;
    bm.x = Bcol[(size_t)(kk + koff) * H];
    bm.y = Bcol[(size_t)(kk + koff + 1) * H];
    c = __builtin_amdgcn_wmma_f32_16x16x4_f32(
        /*neg_a=*/false, a, /*neg_b=*/false, bm,
        /*c_mod=*/(short)0, c, /*reuse_a=*/false, /*reuse_b=*/false);
  }

  // C/D layout: VGPR i -> M = i (lanes 0-15) or i+8 (lanes 16-31); N = lane&15
  const int ncol = tn * 16 + lo16;
#pragma unroll
  for (int i = 0; i < 8; ++i) {
    const int m = tm * 16 + i + half * 8;
    v[(size_t)m * H + ncol] = c[i];
  }
}

// ---------------------------------------------------------------------------
// Kernel 2 (bandwidth-dominant, ~512MB streamed):
//   energy[b,s] = sum_h enc[s,b,h] * v[b,h]     (bias dropped: constant in s)
// One wave per (s,b) pair; 8 x float4 loads per lane = 1024 floats per pair.
// enc stream is non-temporal so the hot 256KB v stays resident in L2.
// ---------------------------------------------------------------------------
__global__ void __launch_bounds__(256)
energy_kernel(const float* __restrict__ enc,
              const float* __restrict__ v,
              float* __restrict__ energy) {
  const int lane = threadIdx.x & 31;
  const int wave = threadIdx.x >> 5;
  const int pair = blockIdx.x * 8 + wave;   // 0 .. S*B-1
  const int b = pair & (B - 1);
  const int s = pair >> 6;

  const v4f* e4 = (const v4f*)(enc + ((size_t)s * B + b) * H);
  const v4f* v4 = (const v4f*)(v + (size_t)b * H);

  float acc = 0.f;
#pragma unroll
  for (int i = 0; i < 8; ++i) {
    const int idx = lane + i * 32;
    v4f a = __builtin_nontemporal_load(e4 + idx);  // TH=NT streaming read
    v4f w = v4[idx];
    acc += a.x * w.x + a.y * w.y + a.z * w.z + a.w * w.w;
  }
  // wave32 reduction
#pragma unroll
  for (int m = 16; m >= 1; m >>= 1) acc += __shfl_xor(acc, m, 32);
  if (lane == 0) energy[(size_t)b * S + s] = acc;
}

// ---------------------------------------------------------------------------
// Kernel 3: out[b, 0, s] = softmax_s(energy[b, s]); one block per b.
// ---------------------------------------------------------------------------
__global__ void __launch_bounds__(256)
softmax_kernel(const float* __restrict__ energy, float* __restrict__ out) {
  __shared__ float red[256];
  const int b = blockIdx.x;
  const int t = threadIdx.x;
  const float* row = energy + (size_t)b * S;

  float vals[8];
  float mx = -INFINITY;
#pragma unroll
  for (int i = 0; i < 8; ++i) {
    vals[i] = row[t + i * 256];
    mx = fmaxf(mx, vals[i]);
  }
  red[t] = mx;
  __syncthreads();
  for (int st = 128; st >= 1; st >>= 1) {
    if (t < st) red[t] = fmaxf(red[t], red[t + st]);
    __syncthreads();
  }
  mx = red[0];
  __syncthreads();

  float sum = 0.f;
#pragma unroll
  for (int i = 0; i < 8; ++i) {
    vals[i] = __expf(vals[i] - mx);
    sum += vals[i];
  }
  red[t] = sum;
  __syncthreads();
  for (int st = 128; st >= 1; st >>= 1) {
    if (t < st) red[t] += red[t + st];
    __syncthreads();
  }
  const float inv = 1.0f / red[0];
#pragma unroll
  for (int i = 0; i < 8; ++i) out[(size_t)b * S + t + i * 256] = vals[i] * inv;
}

extern "C" void kernel_launch(void* const* d_in, const int* in_sizes, int n_in,
                              void* d_out, int out_size, void* d_ws, size_t ws_size,
                              hipStream_t stream) {
  const float* hidden = (const float*)d_in[0];   // (B, H)
  const float* enc    = (const float*)d_in[1];   // (S, B, H)
  const float* W      = (const float*)d_in[2];   // (H, H)
  // d_in[3] = bias: provably irrelevant (constant along softmax axis), skipped.
  float* out = (float*)d_out;                    // (B, 1, S) flat = B*S

  float* v      = (float*)d_ws;                  // B*H   = 65536 f32 (256 KB)
  float* energy = (float*)d_ws + (size_t)B * H;  // B*S   = 131072 f32 (512 KB)

  gemm_vW_wmma<<<64, 128, 0, stream>>>(hidden, W, v);
  energy_kernel<<<(S * B) / 8, 256, 0, stream>>>(enc, v, energy);
  softmax_kernel<<<B, 256, 0, stream>>>(energy, out);
}